// KernelSharedTensorRing_78013785964669
// MI455X (gfx1250) — compile-verified
//
#include <hip/hip_runtime.h>
#include <cstdint>

typedef float v2f __attribute__((ext_vector_type(2)));
typedef float v8f __attribute__((ext_vector_type(8)));
typedef unsigned int v4u __attribute__((ext_vector_type(4)));
typedef int v4i __attribute__((ext_vector_type(4)));
typedef int v8i __attribute__((ext_vector_type(8)));

#define FIRST_RANK 64
#define M_DIM      256
#define SECOND_RANK 64
#define R_DIM      8
#define BATCH      512

#define BT 16            // batch tile (WMMA M)
#define KT 16            // second_rank tile (WMMA N)
#define IN_STRIDE 260    // 256 + 4 pad dwords (TDM pad) -> conflict-free ds_load_b64
#define T_STRIDE  1028   // per-batch stride of t tile: 64*16 + 4 pad

__global__ __launch_bounds__(256)
void str_fused_kernel(const float* __restrict__ input,
                      const float* __restrict__ state,
                      const float* __restrict__ weight,
                      float* __restrict__ out)
{
    __shared__ float lds_in[BT * IN_STRIDE]; // 16x256 input tile (+pad)   ~16.6 KB
    __shared__ float lds_t [BT * T_STRIDE];  // t[b][i][k] (+pad)          ~65.8 KB

    const int bt   = blockIdx.x;   // 0..31 batch tiles
    const int kt   = blockIdx.y;   // 0..3  k tiles
    const int tid  = threadIdx.x;
    const int lane = tid & 31;
    const int wave = tid >> 5;     // 8 waves
    const int l16  = lane & 15;
    const int lh   = lane >> 4;    // lane half (0/1)

    // ---------------- Stage 0: TDM tensor load of input tile into LDS ----------
    if (wave == 0) {
        uint64_t gaddr  = (uint64_t)(uintptr_t)(input + (size_t)bt * BT * M_DIM);
        uint32_t ldsoff = (uint32_t)(uintptr_t)(&lds_in[0]);
        v4u g0;
        g0[0] = 1u;                                   // count=1 valid descriptor
        g0[1] = ldsoff;                               // lds_addr (bytes)
        g0[2] = (uint32_t)gaddr;                      // global_addr[31:0]
        g0[3] = (uint32_t)(gaddr >> 32) | 0x80000000u;// global_addr[56:32], type=2
        v8i g1;
        // data_size=4B(code 2), pad_enable=1, pad_interval=7 (256 DW), pad_amount=3 (4 DW)
        g1[0] = (int)((2u << 16) | (1u << 20) | (7u << 22) | (3u << 25));
        g1[1] = (int)(256u << 16);                    // tensor_dim0 = 256
        g1[2] = (int)(16u  << 16);                    // tensor_dim1 = 16
        g1[3] = (int)(256u << 16);                    // tile_dim0 = 256
        g1[4] = 16;                                   // tile_dim1 = 16
        g1[5] = 256;                                  // tensor_dim0_stride = 256
        g1[6] = 0;
        g1[7] = 0;
        v4i gz4 = {0, 0, 0, 0};                       // groups 2/3 unused (2D)
        v8i gz8 = {0, 0, 0, 0, 0, 0, 0, 0};
        __builtin_amdgcn_tensor_load_to_lds(g0, g1, gz4, gz4, gz8, 0);
        __builtin_amdgcn_s_wait_tensorcnt(0);
    }
    __syncthreads();

    // ---------------- Stage 1: t[b,i,k] = sum_j x[b,j] * W[i,j,k] ---------------
    // Wave w owns i in [8w, 8w+8). f32 WMMA 16x16x4: M=batch tile, N=k tile, K=j.
    v8f acc[8];
    #pragma unroll
    for (int u = 0; u < 8; ++u) {
        v8f z = {0.f, 0.f, 0.f, 0.f, 0.f, 0.f, 0.f, 0.f};
        acc[u] = z;
    }

    // B-matrix per-lane stream pointers: W[i][j0 + 2*lh + v][kt*16 + l16]
    const float* wp[8];
    #pragma unroll
    for (int u = 0; u < 8; ++u) {
        const int i = wave * 8 + u;
        wp[u] = weight + ((size_t)(i * M_DIM + lh * 2) * SECOND_RANK + kt * KT + l16);
        __builtin_prefetch(wp[u], 0, 1);              // global_prefetch_b8
    }

    // A-matrix per-lane LDS pointer: x[b = l16][j0 + 2*lh + v]
    const float* ain = &lds_in[l16 * IN_STRIDE + lh * 2];

    for (int s = 0; s < M_DIM / 4; ++s) {             // 64 K-steps of 4
        v2f a = *(const v2f*)ain;                     // ds_load_b64, conflict-free
        ain += 4;
        #pragma unroll
        for (int u = 0; u < 8; ++u) {
            v2f b;
            b.x = wp[u][0];                           // K row 2*lh + 0
            b.y = wp[u][SECOND_RANK];                 // K row 2*lh + 1
            wp[u] += 4 * SECOND_RANK;                 // advance j by 4
            acc[u] = __builtin_amdgcn_wmma_f32_16x16x4_f32(
                         false, a, false, b, (short)0, acc[u], false, false);
        }
    }

    // Spill t tile to LDS. C/D layout: VGPR d, lanes<16 -> M=d, lanes>=16 -> M=d+8.
    #pragma unroll
    for (int u = 0; u < 8; ++u) {
        const int i = wave * 8 + u;
        #pragma unroll
        for (int d = 0; d < 8; ++d) {
            const int b = d + lh * 8;
            lds_t[b * T_STRIDE + i * KT + l16] = acc[u][d];
        }
    }
    __syncthreads();

    // ---------------- Stage 2: out[r,b,k] = sum_i s[r,b,i] * t[b,i,k] -----------
    // 8r x 16b x 16k = 2048 outputs / 256 threads = 8 per thread (one (r,b), 8 k's).
    const int pr = tid >> 1;
    const int r  = pr >> 4;            // 0..7
    const int b  = pr & 15;            // 0..15
    const int k0 = (tid & 1) * 8;      // 0 or 8

    const float* srow = state + ((size_t)r * BATCH + (size_t)bt * BT + b) * FIRST_RANK;
    const float* trow = &lds_t[b * T_STRIDE + k0];

    float accv[8] = {0.f, 0.f, 0.f, 0.f, 0.f, 0.f, 0.f, 0.f};
    for (int i = 0; i < FIRST_RANK; ++i) {
        const float sv = srow[i];
        #pragma unroll
        for (int kk = 0; kk < 8; ++kk)
            accv[kk] = fmaf(sv, trow[kk], accv[kk]);
        trow += KT;
    }

    float* op = out + ((size_t)r * BATCH + (size_t)bt * BT + b) * SECOND_RANK
                    + kt * KT + k0;
    #pragma unroll
    for (int kk = 0; kk < 8; ++kk)
        op[kk] = accv[kk];
}

extern "C" void kernel_launch(void* const* d_in, const int* in_sizes, int n_in,
                              void* d_out, int out_size, void* d_ws, size_t ws_size,
                              hipStream_t stream) {
    const float* input  = (const float*)d_in[0];   // [512, 256]
    const float* state  = (const float*)d_in[1];   // [8, 512, 64]
    const float* weight = (const float*)d_in[2];   // [64, 256, 64]
    float* out = (float*)d_out;                    // [8, 512, 64]

    dim3 grid(BATCH / BT, SECOND_RANK / KT);       // 32 x 4 = 128 workgroups
    str_fused_kernel<<<grid, 256, 0, stream>>>(input, state, weight, out);
}